// GNNEncoder_30331059044708
// MI455X (gfx1250) — compile-verified
//
#include <hip/hip_runtime.h>
#include <hip/hip_bf16.h>

typedef __attribute__((ext_vector_type(2))) float v2f;
typedef __attribute__((ext_vector_type(8))) float v8f;
typedef int v4i_vs __attribute__((vector_size(16)));   // matches builtin's pointee type

#define H_DIM 128
#define LDS_STRIDE 132   // 128 + 4 floats pad: keeps 16B alignment, breaks 256B bank period

// ---------------------------------------------------------------------------
// CDNA5 async global->LDS copy (16B per lane), ASYNCcnt-tracked.
// ---------------------------------------------------------------------------
__device__ __forceinline__ void cp_async16(float* lds, const float* g) {
#if __has_builtin(__builtin_amdgcn_global_load_async_to_lds_b128)
  __builtin_amdgcn_global_load_async_to_lds_b128(
      (__attribute__((address_space(1))) v4i_vs*)g,
      (__attribute__((address_space(3))) v4i_vs*)lds, 0, 0);
#else
  asm volatile("global_load_async_to_lds_b128 %0, %1, off"
               :: "v"((unsigned)(uintptr_t)lds),
                  "v"((unsigned long long)(uintptr_t)g)
               : "memory");
#endif
}

__device__ __forceinline__ void async_wait0() {
#if __has_builtin(__builtin_amdgcn_s_wait_asynccnt)
  __builtin_amdgcn_s_wait_asynccnt(0);
#else
  asm volatile("s_wait_asynccnt 0x0" ::: "memory");
#endif
}

// D = A(16x4, f32) * B(4x16, f32) + C(16x16, f32) ; full fp32 WMMA on CDNA5
__device__ __forceinline__ v8f wmma4(v8f c, v2f a, v2f b) {
  return __builtin_amdgcn_wmma_f32_16x16x4_f32(
      /*neg_a=*/false, a, /*neg_b=*/false, b,
      /*c_mod=*/(short)0, c, /*reuse_a=*/false, /*reuse_b=*/false);
}

// ---------------------------------------------------------------------------
// Out[M x 128] = A[M x K] @ W^T + bias ; W rows are length-K, row stride wstride.
// Block = 128 threads (4 waves). All waves share one 16-column B panel staged
// into LDS via async copy; each wave owns one 16x16 output tile.
// ---------------------------------------------------------------------------
__global__ void gemm_bias_wmma(const float* __restrict__ A,
                               const float* __restrict__ W,
                               const float* __restrict__ bias,
                               float* __restrict__ Out,
                               int M, int K, int wstride) {
  __shared__ float wtile[16][LDS_STRIDE];

  const int tid     = threadIdx.x;
  const int lane    = tid & 31;
  const int waveId  = tid >> 5;
  const int mTile   = (blockIdx.x * 4 + waveId) * 16;
  const int nTile   = blockIdx.y * 16;
  const int lane16  = lane & 15;
  const int halfSel = lane >> 4;       // 0: K-pair {0,1}, 1: K-pair {2,3}
  const int kb      = halfSel * 2;
  const int col     = nTile + lane16;  // output column (0..127)

  // --- stage B panel (rows nTile..nTile+15 of W) into LDS, 16B async chunks ---
  const int chunksPerRow = K >> 2;               // float4 chunks per row
  for (int cidx = tid; cidx < 16 * chunksPerRow; cidx += 128) {
    const int r = cidx / chunksPerRow;
    const int c = (cidx - r * chunksPerRow) * 4;
    cp_async16(&wtile[r][c], W + (size_t)(nTile + r) * wstride + c);
  }
  async_wait0();
  __syncthreads();

  v8f c;
  const float bv = bias ? bias[col] : 0.0f;
#pragma unroll
  for (int i = 0; i < 8; ++i) c[i] = bv;

  int rowA = mTile + lane16;
  if (rowA > M - 1) rowA = M - 1;      // clamp loads; stores are guarded below
  const float* __restrict__ arow = A + (size_t)rowA * K;
  const float* __restrict__ wrow = &wtile[lane16][0];

  for (int k = 0; k < K; k += 4) {
    v2f a = *(const v2f*)(arow + k + kb);   // global: A[row][k+kb..+1]
    v2f b = *(const v2f*)(wrow + k + kb);   // LDS:    W[col][k+kb..+1]
    c = wmma4(c, a, b);
  }

#pragma unroll
  for (int i = 0; i < 8; ++i) {
    const int m = mTile + (halfSel << 3) + i;   // C layout: lanes 16-31 hold M=8..15
    if (m < M) Out[(size_t)m * H_DIM + col] = c[i];
  }
}

// agg[dst[e]] += hm[src[e]] + emsg[e]  (1 wave per edge, float4 per lane)
__global__ void scatter_add(const float* __restrict__ hm,
                            const float* __restrict__ emsg,
                            const int* __restrict__ src,
                            const int* __restrict__ dst,
                            float* __restrict__ agg, int E) {
  const long long t    = (long long)blockIdx.x * blockDim.x + threadIdx.x;
  const long long edge = t >> 5;
  if (edge >= E) return;
  const int off = ((int)t & 31) * 4;
  const int s = src[edge];
  const int d = dst[edge];
  const float4 hv = *(const float4*)(hm + (size_t)s * H_DIM + off);
  const float4 ev = *(const float4*)(emsg + (size_t)edge * H_DIM + off);
  float* base = agg + (size_t)d * H_DIM + off;
  unsafeAtomicAdd(base + 0, hv.x + ev.x);
  unsafeAtomicAdd(base + 1, hv.y + ev.y);
  unsafeAtomicAdd(base + 2, hv.z + ev.z);
  unsafeAtomicAdd(base + 3, hv.w + ev.w);
}

__global__ void zero_f32v4(float4* __restrict__ p, long long n4) {
  const long long i = (long long)blockIdx.x * blockDim.x + threadIdx.x;
  if (i < n4) p[i] = make_float4(0.f, 0.f, 0.f, 0.f);
}

__device__ __forceinline__ float sigmoidf_(float x) {
  return 1.0f / (1.0f + __expf(-x));
}

// ---------------------------------------------------------------------------
// Fused GRU cell: gi = agg@W_ih^T + b_ih, gh = h@W_hh^T + b_hh, gates in the
// C fragments. Six 16x128 B panels (r/z/n for W_ih and W_hh) async-staged to
// LDS and shared by all 4 waves of the block. One wave per 16x16 h_out tile.
// ---------------------------------------------------------------------------
__global__ void gru_wmma(const float* __restrict__ agg, const float* __restrict__ h,
                         const float* __restrict__ W_ih, const float* __restrict__ W_hh,
                         const float* __restrict__ b_ih, const float* __restrict__ b_hh,
                         float* __restrict__ h_out, int M) {
  __shared__ float wt[6][16][LDS_STRIDE];   // ~49.5 KB of the 320 KB WGP LDS

  const int tid     = threadIdx.x;
  const int lane    = tid & 31;
  const int waveId  = tid >> 5;
  const int mTile   = (blockIdx.x * 4 + waveId) * 16;
  const int nTile   = blockIdx.y * 16;
  const int lane16  = lane & 15;
  const int halfSel = lane >> 4;
  const int kb      = halfSel * 2;
  const int col     = nTile + lane16;

  // --- stage 6 B panels: p<3 from W_ih (r,z,n), p>=3 from W_hh (r,z,n) ---
  for (int cidx = tid; cidx < 6 * 16 * 32; cidx += 128) {
    const int p   = cidx >> 9;            // /512
    const int rem = cidx & 511;
    const int r   = rem >> 5;             // /32
    const int cc  = (rem & 31) * 4;
    const float* Wsel = (p < 3) ? W_ih : W_hh;
    const int rowBase = nTile + ((p < 3) ? p : (p - 3)) * 128;
    cp_async16(&wt[p][r][cc], Wsel + (size_t)(rowBase + r) * H_DIM + cc);
  }
  async_wait0();
  __syncthreads();

  v8f cir, ciz, cin, chr_, chz, chn;
#pragma unroll
  for (int i = 0; i < 8; ++i) {
    cir[i]  = b_ih[col];        chr_[i] = b_hh[col];
    ciz[i]  = b_ih[col + 128];  chz[i]  = b_hh[col + 128];
    cin[i]  = b_ih[col + 256];  chn[i]  = b_hh[col + 256];
  }

  int rowA = mTile + lane16;
  if (rowA > M - 1) rowA = M - 1;
  const float* __restrict__ arow = agg + (size_t)rowA * H_DIM;
  const float* __restrict__ hrow = h   + (size_t)rowA * H_DIM;
  const float* __restrict__ wir = &wt[0][lane16][0];
  const float* __restrict__ wiz = &wt[1][lane16][0];
  const float* __restrict__ win = &wt[2][lane16][0];
  const float* __restrict__ whr = &wt[3][lane16][0];
  const float* __restrict__ whz = &wt[4][lane16][0];
  const float* __restrict__ whn = &wt[5][lane16][0];

  for (int k = 0; k < H_DIM; k += 4) {
    const v2f ax = *(const v2f*)(arow + k + kb);
    const v2f ah = *(const v2f*)(hrow + k + kb);
    cir  = wmma4(cir,  ax, *(const v2f*)(wir + k + kb));
    ciz  = wmma4(ciz,  ax, *(const v2f*)(wiz + k + kb));
    cin  = wmma4(cin,  ax, *(const v2f*)(win + k + kb));
    chr_ = wmma4(chr_, ah, *(const v2f*)(whr + k + kb));
    chz  = wmma4(chz,  ah, *(const v2f*)(whz + k + kb));
    chn  = wmma4(chn,  ah, *(const v2f*)(whn + k + kb));
  }

#pragma unroll
  for (int i = 0; i < 8; ++i) {
    const int m = mTile + (halfSel << 3) + i;
    const float r  = sigmoidf_(cir[i] + chr_[i]);
    const float z  = sigmoidf_(ciz[i] + chz[i]);
    const float nn = tanhf(cin[i] + r * chn[i]);
    if (m < M) {
      const float hv = h[(size_t)m * H_DIM + col];
      h_out[(size_t)m * H_DIM + col] = (1.0f - z) * nn + z * hv;
    }
  }
}

extern "C" void kernel_launch(void* const* d_in, const int* in_sizes, int n_in,
                              void* d_out, int out_size, void* d_ws, size_t ws_size,
                              hipStream_t stream) {
  const float* node_feat  = (const float*)d_in[0];
  const int*   edge_index = (const int*)  d_in[1];
  const float* edge_feat  = (const float*)d_in[2];
  const float* W_node = (const float*)d_in[3];
  const float* b_node = (const float*)d_in[4];
  const float* W_edge = (const float*)d_in[5];
  const float* b_edge = (const float*)d_in[6];
  const float* W_msg  = (const float*)d_in[7];   // [128 x 256]
  const float* b_msg  = (const float*)d_in[8];
  const float* W_ih   = (const float*)d_in[9];   // [384 x 128]
  const float* W_hh   = (const float*)d_in[10];
  const float* b_ih   = (const float*)d_in[11];
  const float* b_hh   = (const float*)d_in[12];

  const int N = in_sizes[0] / 64;   // NODE_DIM = 64
  const int E = in_sizes[2] / 32;   // EDGE_DIM = 32

  float* h_out = (float*)d_out;                          // N x 128
  float* e_out = (float*)d_out + (size_t)N * H_DIM;      // E x 128 (output `e`)

  float* ws   = (float*)d_ws;
  float* emsg = ws;                                      // E x 128
  float* hm   = emsg + (size_t)E * H_DIM;                // N x 128
  float* aggb = hm   + (size_t)N * H_DIM;                // N x 128
  float* hA   = aggb + (size_t)N * H_DIM;                // N x 128
  float* hB   = hA   + (size_t)N * H_DIM;                // N x 128

  const dim3 blk(128);
  auto gemmGrid = [](int M) { return dim3((unsigned)((M + 63) / 64), 8); };

  // h0 = node_feat @ W_node^T + b_node
  gemm_bias_wmma<<<gemmGrid(N), blk, 0, stream>>>(node_feat, W_node, b_node, hA, N, 64, 64);
  // e = edge_feat @ W_edge^T + b_edge  (written straight into output slot)
  gemm_bias_wmma<<<gemmGrid(E), blk, 0, stream>>>(edge_feat, W_edge, b_edge, e_out, E, 32, 32);
  // e_msg = e @ W_msg_e^T + b_msg  (W_msg_e = W_msg[:, 128:], row stride 256)
  gemm_bias_wmma<<<gemmGrid(E), blk, 0, stream>>>(e_out, W_msg + H_DIM, b_msg, emsg, E, 128, 256);

  const float* h_cur = hA;
  for (int it = 0; it < 3; ++it) {
    float* h_next = (it == 2) ? h_out : ((it == 0) ? hB : hA);

    // hm = h @ W_msg_h^T  (hoisted out of the edge dimension; no bias)
    gemm_bias_wmma<<<gemmGrid(N), blk, 0, stream>>>(h_cur, W_msg, nullptr, hm, N, 128, 256);

    const long long n4 = (long long)N * H_DIM / 4;
    zero_f32v4<<<(unsigned)((n4 + 255) / 256), 256, 0, stream>>>((float4*)aggb, n4);

    // agg[dst] += hm[src] + e_msg  (== segment_sum of m)
    const long long sth = (long long)E * 32;
    scatter_add<<<(unsigned)((sth + 255) / 256), 256, 0, stream>>>(
        hm, emsg, edge_index, edge_index + E, aggb, E);

    // h = GRU(agg, h)
    gru_wmma<<<gemmGrid(N), blk, 0, stream>>>(aggb, h_cur, W_ih, W_hh, b_ih, b_hh, h_next, N);
    h_cur = h_next;
  }
}